// MatrixNetwork_10599979286658
// MI455X (gfx1250) — compile-verified
//
#include <hip/hip_runtime.h>

// CDNA5 / gfx1250: wave32, WMMA (not MFMA), async global->LDS copies.
typedef __attribute__((ext_vector_type(2))) float v2f;
typedef __attribute__((ext_vector_type(8))) float v8f;
typedef __attribute__((ext_vector_type(2))) int   v2i;

#define AS1 __attribute__((address_space(1)))
#define AS3 __attribute__((address_space(3)))

#if defined(__has_builtin)
# if __has_builtin(__builtin_amdgcn_global_load_async_to_lds_b64)
#  define ASYNC_PATH 1
# endif
#endif
#ifndef ASYNC_PATH
# define ASYNC_PATH 0
#endif

#define DIM        64
#define NCOLS      8192
#define PATH_LEN   256
// LDS layout: element pair (row, 2cp),(row, 2cp+1) at dword  cp*CP_STRIDE + 2*row.
// CP_STRIDE = 128 + 8 pad: A-operand ds_load_b64 hits all 64 banks exactly once
// (lanes0-15: banks 2m+c; lanes16-31: cp+=4 -> +544 dwords = +32 banks).
#define CP_STRIDE  136
#define LDS_DWORDS (32 * CP_STRIDE)
#define WAVES_PER_WG   8
#define THREADS_PER_WG 256                   // 8 waves * 16 cols = 128 cols/WG
#define PAIRS_PER_THREAD (DIM * DIM / 2 / THREADS_PER_WG)   // 8

__device__ __forceinline__ void wait_async_zero() {
#if ASYNC_PATH
# if __has_builtin(__builtin_amdgcn_s_wait_asynccnt)
    __builtin_amdgcn_s_wait_asynccnt(0);
# else
    asm volatile("s_wait_asynccnt 0" ::: "memory");
# endif
#endif
}

// Bijective map of pair index u (0..2047) -> (cp 0..31, row 0..63) such that the
// 32 lanes of any wave target 32 distinct LDS bank-pairs (conflict-free b64)
// while global reads stay in 64B-contiguous chunks.
__device__ __forceinline__ void pair_coords(int u, int& cp, int& row) {
    cp  = ((u >> 2) & 7) | (((u >> 5) & 3) << 3);
    row = (u & 3)        | ((u >> 7) << 2);
}

__device__ __forceinline__ void async_copy_pair(const float* src, float* dst) {
#if ASYNC_PATH
    __builtin_amdgcn_global_load_async_to_lds_b64(
        (AS1 v2i*)(uintptr_t)src, (AS3 v2i*)(uintptr_t)dst, 0, 0);
#endif
}

// x stays in registers in WMMA C/D layout for the whole 256-step chain.
// acc[kb] (v8f): VGPR v holds x[row = kb*16 + v + (lane<16 ? 0 : 8)][col0 + lane%16].
// Fed back as the B operand: VGPR pair (2p,2p+1) of acc[kb] supplies K-rows
// {kb*16+2p, +1, +8, +9}; the A operand loads exactly those 4 columns of M
// (one ds_load_b64 per lane) from the pair-layout LDS image.
// Each product chain starts from an inline-zero accumulator; the "+x" term is
// folded into the normalization epilogue (no acc->D register copies).
__global__ __launch_bounds__(THREADS_PER_WG)
void matnet_path_kernel(const float* __restrict__ x,
                        const int*   __restrict__ edge_ids,
                        const float* __restrict__ edge_matrix,
                        const float* __restrict__ scale,
                        float*       __restrict__ out)
{
    __shared__ float ldsM[2][LDS_DWORDS];    // double buffered, 2 x 17 KB

    const int tid  = threadIdx.x;
    const int wave = tid >> 5;               // warpSize == 32 on gfx1250
    const int lane = tid & 31;
    const int m    = lane & 15;              // A row-in-block / B,C,D column
    const int hi   = lane >> 4;              // lane half
    const int col0 = (blockIdx.x * WAVES_PER_WG + wave) * 16;
    const int mycol = col0 + m;

    // ---- load x tile into C/D-layout accumulators (once) ----
    v8f acc[4];
#pragma unroll
    for (int kb = 0; kb < 4; ++kb)
#pragma unroll
        for (int v = 0; v < 8; ++v)
            acc[kb][v] = x[(kb * 16 + v + hi * 8) * NCOLS + mycol];

    // ---- stage M[edge 0] into buffer 0 ----
    {
        const float* M0 = edge_matrix + (size_t)edge_ids[0] * (DIM * DIM);
        float* dst0 = &ldsM[0][0];
#pragma unroll
        for (int k = 0; k < PAIRS_PER_THREAD; ++k) {
            int cp, row; pair_coords(k * THREADS_PER_WG + tid, cp, row);
            const float* src = M0 + row * DIM + 2 * cp;
            float*       dst = dst0 + cp * CP_STRIDE + 2 * row;
#if ASYNC_PATH
            async_copy_pair(src, dst);
#else
            *(v2f*)dst = *(const v2f*)src;
#endif
        }
    }
    wait_async_zero();
    __syncthreads();

    for (int s = 0; s < PATH_LEN; ++s) {
        const float* lds = &ldsM[s & 1][0];
        const bool have_next = (s + 1 < PATH_LEN);

        // 1) start staging next M into the other buffer (overlaps the WMMAs)
#if !ASYNC_PATH
        v2f stage[PAIRS_PER_THREAD];
#endif
        if (have_next) {
            const float* Mn = edge_matrix + (size_t)edge_ids[s + 1] * (DIM * DIM);
#if ASYNC_PATH
            float* dstb = &ldsM[(s + 1) & 1][0];
#endif
#pragma unroll
            for (int k = 0; k < PAIRS_PER_THREAD; ++k) {
                int cp, row; pair_coords(k * THREADS_PER_WG + tid, cp, row);
                const float* src = Mn + row * DIM + 2 * cp;
#if ASYNC_PATH
                async_copy_pair(src, dstb + cp * CP_STRIDE + 2 * row);
#else
                stage[k] = *(const v2f*)src;
#endif
            }
        }

        // 2) y = M @ x : 64 x V_WMMA_F32_16X16X4_F32 per wave, zero-init C,
        //    every A operand a single aligned ds_load_b64 (immediate offsets).
        const float* abase = lds + hi * (4 * CP_STRIDE) + 2 * m;
        v8f ynew[4];
#pragma unroll
        for (int ib = 0; ib < 4; ++ib) {
            v8f c = {};                       // SRC2 = inline 0 (hoistable)
#pragma unroll
            for (int kb = 0; kb < 4; ++kb) {
#pragma unroll
                for (int p = 0; p < 4; ++p) {
                    v2f a = *(const v2f*)(abase + (kb * 8 + p) * CP_STRIDE + ib * 32);
                    v2f b;
                    b[0] = acc[kb][2 * p];
                    b[1] = acc[kb][2 * p + 1];
                    c = __builtin_amdgcn_wmma_f32_16x16x4_f32(
                            false, a, false, b, (short)0, c, false, false);
                }
            }
            ynew[ib] = c;
        }

        // 3) y += x fused with the per-column L2 norm (all 64 rows in-wave)
        float ss = 0.0f;
#pragma unroll
        for (int ib = 0; ib < 4; ++ib)
#pragma unroll
            for (int v = 0; v < 8; ++v) {
                float t = ynew[ib][v] + acc[ib][v];
                ynew[ib][v] = t;
                ss = fmaf(t, t, ss);
            }
        ss += __shfl_xor(ss, 16, 32);
        const float inv = 1.0f / sqrtf(ss);
#pragma unroll
        for (int ib = 0; ib < 4; ++ib)
#pragma unroll
            for (int v = 0; v < 8; ++v)
                acc[ib][v] = ynew[ib][v] * inv;

        // 4) commit staged M (fallback path), fence asyncs, one barrier per step
#if !ASYNC_PATH
        if (have_next) {
            float* dstb = &ldsM[(s + 1) & 1][0];
#pragma unroll
            for (int k = 0; k < PAIRS_PER_THREAD; ++k) {
                int cp, row; pair_coords(k * THREADS_PER_WG + tid, cp, row);
                *(v2f*)(dstb + cp * CP_STRIDE + 2 * row) = stage[k];
            }
        }
#endif
        wait_async_zero();
        __syncthreads();
    }

    // ---- epilogue: out = scale * x ----
    const float sc = scale[0];
#pragma unroll
    for (int kb = 0; kb < 4; ++kb)
#pragma unroll
        for (int v = 0; v < 8; ++v)
            out[(kb * 16 + v + hi * 8) * NCOLS + mycol] = acc[kb][v] * sc;
}

extern "C" void kernel_launch(void* const* d_in, const int* in_sizes, int n_in,
                              void* d_out, int out_size, void* d_ws, size_t ws_size,
                              hipStream_t stream) {
    const float* x           = (const float*)d_in[0];
    const int*   edge_ids    = (const int*)  d_in[1];
    const float* edge_matrix = (const float*)d_in[2];
    const float* scale       = (const float*)d_in[3];
    float* out = (float*)d_out;

    dim3 grid(NCOLS / (WAVES_PER_WG * 16));  // 64 workgroups
    dim3 block(THREADS_PER_WG);              // 8 waves -> 8192 columns total
    hipLaunchKernelGGL(matnet_path_kernel, grid, block, 0, stream,
                       x, edge_ids, edge_matrix, scale, out);
}